// LSEPLoss_28544352649245
// MI455X (gfx1250) — compile-verified
//
#include <hip/hip_runtime.h>
#include <hip/hip_bf16.h>
#include <math.h>

typedef __attribute__((ext_vector_type(2))) float v2f;
typedef __attribute__((ext_vector_type(8))) float v8f;

// Each wave handles 16 rows of C=256 columns.
// Lane l (l&15 = r) streams row base+r; lanes 0-15 take cols 8k+0..3,
// lanes 16-31 take cols 8k+4..7.  Per-lane partials s0,s1 are then
// reduced across the half-wave pair with V_WMMA_F32_16X16X4_F32
// (A = 16x4 partials, B = ones) -> row sums land per-row in C VGPRs.
__global__ __launch_bounds__(256) void lsep_rows_kernel(
    const float* __restrict__ logit, const int* __restrict__ target,
    float* __restrict__ row_out) {
  const int lane = threadIdx.x & 31;
  const int wave = threadIdx.x >> 5;
  const int rowBase = (blockIdx.x * 8 + wave) * 16;
  const int r = lane & 15;
  const int row = rowBase + r;
  const int colOff = (lane & 16) ? 4 : 0;

  const float4* lp = (const float4*)(logit + (size_t)row * 256 + colOff);
  const int4* tp = (const int4*)(target + (size_t)row * 256 + colOff);

  float s0 = 0.0f;  // sum over t==0 of exp(x)
  float s1 = 0.0f;  // sum over t==1 of exp(-x)

#pragma unroll 4
  for (int k = 0; k < 32; ++k) {
    float4 x4 = lp[2 * k];  // cols colOff + 8k .. +3
    int4 t4 = tp[2 * k];
    float xs[4] = {x4.x, x4.y, x4.z, x4.w};
    int ts[4] = {t4.x, t4.y, t4.z, t4.w};
#pragma unroll
    for (int j = 0; j < 4; ++j) {
      float x = xs[j];
      // NaN/Inf -> 0 (NaN compares false; |Inf| < Inf is false)
      if (!(fabsf(x) < __builtin_inff())) x = 0.0f;
      const int t = ts[j];
      const float e = __expf(t ? -x : x);  // one transcendental per element
      s0 += t ? 0.0f : e;
      s1 += t ? e : 0.0f;
    }
  }

  // Cross-half-wave reduction via WMMA:
  // A layout (f32 16x4): lanes 0-15 supply (M=r, K=0)=VGPR0, (K=1)=VGPR1;
  // lanes 16-31 supply (M=r, K=2), (K=3).  B = ones(4x16), C = 0.
  // D[m,n] = s_lo(m) + s_hi(m) = full row sum, replicated over n.
  v2f A0;
  A0.x = s0;
  A0.y = 0.0f;
  v2f A1;
  A1.x = s1;
  A1.y = 0.0f;
  v2f Bones;
  Bones.x = 1.0f;
  Bones.y = 1.0f;
  v8f C0 = {};
  v8f C1 = {};
  C0 = __builtin_amdgcn_wmma_f32_16x16x4_f32(false, A0, false, Bones,
                                             (short)0, C0, false, false);
  C1 = __builtin_amdgcn_wmma_f32_16x16x4_f32(false, A1, false, Bones,
                                             (short)0, C1, false, false);

  // C/D layout: VGPR q holds row q (lanes 0-15, N=lane) or row 8+q
  // (lanes 16-31, N=lane-16).  Lane 0 -> rows base+0..7, lane 16 -> base+8..15.
  if ((lane & 15) == 0) {
    const int base = rowBase + ((lane & 16) ? 8 : 0);
#pragma unroll
    for (int q = 0; q < 8; ++q) {
      row_out[base + q] = log1pf(C0[q] * C1[q]);
    }
  }
}

// Deterministic fixed-tree mean over N row losses.
__global__ __launch_bounds__(256) void lsep_reduce_kernel(
    const float* __restrict__ row_out, float* __restrict__ out, int n) {
  __shared__ float sdata[256];
  float s = 0.0f;
  for (int i = threadIdx.x; i < n; i += 256) s += row_out[i];
  sdata[threadIdx.x] = s;
  __syncthreads();
#pragma unroll
  for (int off = 128; off > 0; off >>= 1) {
    if ((int)threadIdx.x < off) sdata[threadIdx.x] += sdata[threadIdx.x + off];
    __syncthreads();
  }
  if (threadIdx.x == 0) out[0] = sdata[0] * (1.0f / (float)n);
}

extern "C" void kernel_launch(void* const* d_in, const int* in_sizes, int n_in,
                              void* d_out, int out_size, void* d_ws,
                              size_t ws_size, hipStream_t stream) {
  const float* logit = (const float*)d_in[0];
  const int* target = (const int*)d_in[1];
  const int C = 256;
  const int N = in_sizes[0] / C;  // 4096
  float* rows = (float*)d_ws;     // N floats of scratch (16 KB)

  const int rowsPerBlock = 8 * 16;  // 8 waves x 16 rows
  lsep_rows_kernel<<<N / rowsPerBlock, 256, 0, stream>>>(logit, target, rows);
  lsep_reduce_kernel<<<1, 256, 0, stream>>>(rows, (float*)d_out, N);
}